// GNN_55207509622967
// MI455X (gfx1250) — compile-verified
//
#include <hip/hip_runtime.h>
#include <math.h>

typedef float v2f __attribute__((ext_vector_type(2)));
typedef float v8f __attribute__((ext_vector_type(8)));

#define F_IN 128
#define HDIM 128
#define NEG_SLOPE 0.2f

// ---- monotonic uint key for float atomic-max (handles negatives) ----
__device__ __forceinline__ unsigned fkey(float f) {
    int fi = __float_as_int(f);
    return (unsigned)(fi ^ ((fi >> 31) | 0x80000000));
}
__device__ __forceinline__ float funkey(unsigned k) {
    int fi = (k & 0x80000000u) ? (int)(k ^ 0x80000000u) : (int)(~k);
    return __int_as_float(fi);
}
#define KEY_NEG_INF 0x007FFFFFu  // fkey(-inf)

// =====================================================================
// Kernel 1: h = x @ W via V_WMMA_F32_16X16X4_F32, fused a_s/a_d dots.
// One block = 16 rows of x, 8 waves each own a 16-wide column block.
// =====================================================================
__global__ __launch_bounds__(256)
void gat_gemm(const float* __restrict__ x, const float* __restrict__ W,
              const float* __restrict__ att_src, const float* __restrict__ att_dst,
              float* __restrict__ h, float* __restrict__ a_s, float* __restrict__ a_d,
              int N) {
    const int XS = F_IN + 4;            // pad: stride 132 -> conflict-free LDS
    __shared__ float xs[16 * XS];
    __shared__ float hs[16 * XS];
    __shared__ float ps[16 * 16];
    __shared__ float pd[16 * 16];

    const int tid = threadIdx.x;
    const int M0  = blockIdx.x * 16;

    // stage 16x128 x-tile into LDS (clamped loads keep EXEC full)
    for (int i = tid; i < 16 * F_IN; i += 256) {
        int r = i >> 7, k = i & 127;
        int row = M0 + r; if (row >= N) row = N - 1;
        xs[r * XS + k] = x[(size_t)row * F_IN + k];
    }
    __syncthreads();

    const int lane = tid & 31;
    const int wave = tid >> 5;
    const int col0 = wave * 16;
    const int m    = lane & 15;
    const int hi   = lane >> 4;   // 0: lanes 0-15, 1: lanes 16-31
    const int kk   = hi * 2;      // K sub-offset per ISA A/B layout

    v8f c = {0.f, 0.f, 0.f, 0.f, 0.f, 0.f, 0.f, 0.f};
    for (int k0 = 0; k0 < F_IN; k0 += 4) {
        v2f a, b;
        // A 16x4: v0 = K{0|2}, v1 = K{1|3} per lane-half
        a.x = xs[m * XS + k0 + kk];
        a.y = xs[m * XS + k0 + kk + 1];
        // B 4x16: lanes0-15 rows K=k0..k0+1, lanes16-31 rows k0+2..k0+3
        b.x = W[(size_t)(k0 + kk) * HDIM + col0 + m];
        b.y = W[(size_t)(k0 + kk + 1) * HDIM + col0 + m];
        c = __builtin_amdgcn_wmma_f32_16x16x4_f32(false, a, false, b,
                                                  (short)0, c, false, false);
    }

    // scatter D (VGPR r -> rows r / r+8) into LDS h-tile
    {
        int mb = hi * 8;
        #pragma unroll
        for (int r = 0; r < 8; ++r)
            hs[(mb + r) * XS + col0 + m] = c[r];
    }
    __syncthreads();

    // write h tile, coalesced
    for (int i = tid; i < 16 * HDIM; i += 256) {
        int r = i >> 7, k = i & 127;
        int row = M0 + r;
        if (row < N) h[(size_t)row * HDIM + k] = hs[r * XS + k];
    }

    // fused attention-coefficient dots: a_s = h.att_src, a_d = h.att_dst
    {
        int r = tid >> 4, g = tid & 15;
        float ss = 0.f, sd = 0.f;
        #pragma unroll
        for (int t = 0; t < 8; ++t) {
            int j = g + t * 16;
            float v = hs[r * XS + j];
            ss += v * att_src[j];
            sd += v * att_dst[j];
        }
        ps[r * 16 + g] = ss;
        pd[r * 16 + g] = sd;
    }
    __syncthreads();
    if (tid < 16) {
        float ss = 0.f, sd = 0.f;
        #pragma unroll
        for (int g = 0; g < 16; ++g) { ss += ps[tid * 16 + g]; sd += pd[tid * 16 + g]; }
        int row = M0 + tid;
        if (row < N) { a_s[row] = ss; a_d[row] = sd; }
    }
}

// =====================================================================
// Kernel 2: init per-node softmax state
// =====================================================================
__global__ void gat_init(unsigned* __restrict__ emax, float* __restrict__ denom, int N) {
    int i = blockIdx.x * blockDim.x + threadIdx.x;
    if (i < N) { emax[i] = KEY_NEG_INF; denom[i] = 0.f; }
}

// =====================================================================
// Kernel 3: edge scores (leaky-relu) + segment max via uint atomicMax
// =====================================================================
__global__ __launch_bounds__(256)
void gat_scores(const int* __restrict__ ei, const float* __restrict__ a_s,
                const float* __restrict__ a_d, float* __restrict__ score,
                unsigned* __restrict__ emax, int E, int N) {
    int e = blockIdx.x * blockDim.x + threadIdx.x;
    if (e >= E + N) return;
    int s, d;
    if (e < E) { s = ei[e]; d = ei[E + e]; } else { s = d = e - E; }
    float sc = a_s[s] + a_d[d];
    sc = sc > 0.f ? sc : NEG_SLOPE * sc;
    score[e] = sc;
    atomicMax(&emax[d], fkey(sc));
}

// =====================================================================
// Kernel 4: ex = exp(score - max[dst]) (in place) + segment sum
// =====================================================================
__global__ __launch_bounds__(256)
void gat_expsum(const int* __restrict__ ei, float* __restrict__ score,
                const unsigned* __restrict__ emax, float* __restrict__ denom,
                int E, int N) {
    int e = blockIdx.x * blockDim.x + threadIdx.x;
    if (e >= E + N) return;
    int d = (e < E) ? ei[E + e] : (e - E);
    float ex = __expf(score[e] - funkey(emax[d]));
    score[e] = ex;
    atomicAdd(&denom[d], ex);
}

// =====================================================================
// Kernel 5: agg[dst] += alpha * h[src] -- one wave32 per edge,
// float4 per lane (coalesced 512B gather, L2-resident scatter)
// =====================================================================
__global__ __launch_bounds__(256)
void gat_aggregate(const int* __restrict__ ei, const float* __restrict__ ex,
                   const float* __restrict__ denom, const float* __restrict__ h,
                   float* __restrict__ agg, int E, int N) {
    const int lane = threadIdx.x & 31;
    long long wid = (long long)blockIdx.x * 8 + (threadIdx.x >> 5);
    if (wid >= (long long)(E + N)) return;
    int e = (int)wid;
    int s, d;
    if (e < E) { s = ei[e]; d = ei[E + e]; } else { s = d = e - E; }
    float alpha = ex[e] / denom[d];
    float4 h4 = ((const float4*)(h + (size_t)s * HDIM))[lane];
    float* ap = agg + (size_t)d * HDIM + lane * 4;
    atomicAdd(ap + 0, alpha * h4.x);
    atomicAdd(ap + 1, alpha * h4.y);
    atomicAdd(ap + 2, alpha * h4.z);
    atomicAdd(ap + 3, alpha * h4.w);
}

// =====================================================================
// Kernel 6: in-place SiLU(agg + bias) -- one wave per node
// =====================================================================
__global__ __launch_bounds__(256)
void gat_silu(float* __restrict__ agg, const float* __restrict__ bias, int N) {
    const int lane = threadIdx.x & 31;
    long long wid = (long long)blockIdx.x * 8 + (threadIdx.x >> 5);
    if (wid >= (long long)N) return;
    float4* p = (float4*)(agg + (size_t)wid * HDIM) + lane;
    float4 v = *p;
    float4 b4 = ((const float4*)bias)[lane];
    v.x += b4.x; v.y += b4.y; v.z += b4.z; v.w += b4.w;
    v.x = v.x / (1.f + __expf(-v.x));
    v.y = v.y / (1.f + __expf(-v.y));
    v.z = v.z / (1.f + __expf(-v.z));
    v.w = v.w / (1.f + __expf(-v.w));
    *p = v;
}

// =====================================================================
// Kernel 7: per-graph mean pool (batch sorted -> binary-search range)
// fused with final linear: out[g] = pooled . fin_w + fin_b
// =====================================================================
__global__ __launch_bounds__(128)
void gat_pool(const float* __restrict__ gnn, const int* __restrict__ batch,
              const float* __restrict__ fin_w, const float* __restrict__ fin_b,
              float* __restrict__ out, int N) {
    __shared__ int s_lo, s_hi;
    __shared__ float red[128];
    const int g = blockIdx.x, j = threadIdx.x;
    if (j == 0) {
        int lo = 0, hb = N;
        while (lo < hb) { int mid = (lo + hb) >> 1; if (batch[mid] < g) lo = mid + 1; else hb = mid; }
        s_lo = lo;
        lo = 0; hb = N;
        while (lo < hb) { int mid = (lo + hb) >> 1; if (batch[mid] < g + 1) lo = mid + 1; else hb = mid; }
        s_hi = lo;
    }
    __syncthreads();
    const int lo = s_lo, hiB = s_hi;
    float acc = 0.f;
    for (int n = lo; n < hiB; ++n) acc += gnn[(size_t)n * HDIM + j];
    float cnt = (float)(hiB - lo); if (cnt < 1.f) cnt = 1.f;
    red[j] = (acc / cnt) * fin_w[j];
    __syncthreads();
    for (int sft = 64; sft > 0; sft >>= 1) {
        if (j < sft) red[j] += red[j + sft];
        __syncthreads();
    }
    if (j == 0) out[g] = red[0] + fin_b[0];
}

// =====================================================================
extern "C" void kernel_launch(void* const* d_in, const int* in_sizes, int n_in,
                              void* d_out, int out_size, void* d_ws, size_t ws_size,
                              hipStream_t stream) {
    const float* x       = (const float*)d_in[0];
    const int*   ei      = (const int*)d_in[1];
    const int*   batch   = (const int*)d_in[2];
    const float* W       = (const float*)d_in[3];
    const float* att_src = (const float*)d_in[4];
    const float* att_dst = (const float*)d_in[5];
    const float* bias    = (const float*)d_in[6];
    const float* fin_w   = (const float*)d_in[7];
    const float* fin_b   = (const float*)d_in[8];

    const int N = in_sizes[0] / F_IN;   // 100000
    const int E = in_sizes[1] / 2;      // 1600000
    const int G = out_size;             // 64
    const int ETOT = E + N;

    // ---- workspace layout (all 16B aligned) ----
    float*    h     = (float*)d_ws;                       // N*H
    float*    a_s   = h + (size_t)N * HDIM;               // N
    float*    a_d   = a_s + N;                            // N
    unsigned* emax  = (unsigned*)(a_d + N);               // N
    float*    denom = (float*)(emax + N);                 // N
    float*    score = denom + N;                          // E+N (ex in place)
    float*    agg   = score + ETOT;                       // N*H (gnn in place)

    hipMemsetAsync(agg, 0, (size_t)N * HDIM * sizeof(float), stream);

    gat_init<<<(N + 255) / 256, 256, 0, stream>>>(emax, denom, N);
    gat_gemm<<<(N + 15) / 16, 256, 0, stream>>>(x, W, att_src, att_dst, h, a_s, a_d, N);
    gat_scores<<<(ETOT + 255) / 256, 256, 0, stream>>>(ei, a_s, a_d, score, emax, E, N);
    gat_expsum<<<(ETOT + 255) / 256, 256, 0, stream>>>(ei, score, emax, denom, E, N);
    gat_aggregate<<<(ETOT + 7) / 8, 256, 0, stream>>>(ei, score, denom, h, agg, E, N);
    gat_silu<<<(N + 7) / 8, 256, 0, stream>>>(agg, bias, N);
    gat_pool<<<G, 128, 0, stream>>>(agg, batch, fin_w, fin_b, (float*)d_out, N);
}